// MultiHeadAttention_50534585205286
// MI455X (gfx1250) — compile-verified
//
#include <hip/hip_runtime.h>
#include <hip/hip_bf16.h>

// ---------------------------------------------------------------------------
// MultiHeadAttention for MI455X (gfx1250), fp32 end-to-end via
// V_WMMA_F32_16X16X4_F32.
//
// Roofline: ~30 GFLOP vs ~0.3 GB mandatory HBM traffic (attn output 268 MB
// written once) => memory-bound. Fused stripe kernel keeps the 16x2048 score
// stripe in LDS (131 KB of the 320 KB/WGP), doing scores -> softmax ->
// top-1 mask -> attn.V without round-tripping attn through HBM (it is
// written to global exactly once, as the required kernel output).
// ---------------------------------------------------------------------------

typedef float v2f __attribute__((ext_vector_type(2)));
typedef float v8f __attribute__((ext_vector_type(8)));

#define LQ_   2048
#define DM_   512
#define NH_   8
#define DK_   64
#define BB_   2
#define ROWS_ (BB_ * LQ_)      // 4096 flattened (b, l) rows

#define STRIPE_LD 2052         // 2048 + 4 pad: lane l16 -> bank (l16*4)%64, conflict-free
#define SMEM_FLOATS (16 * STRIPE_LD + 256 + 1024)
#define SMEM_BYTES  (SMEM_FLOATS * 4)

static __device__ const float INV_TEMPER = 0.044194173824159216f;  // 1/sqrt(512)

// ---- WMMA helpers ----------------------------------------------------------
// f32 16x16x4 fragment layouts (wave32):
//   A (16x4):  lane L -> m = L%16 ; VGPR j holds K = 2*(L/16)+j
//   B (4x16):  lane L -> n = L%16 ; VGPR j holds K = 2*j + L/16
//   C (16x16): lane L -> n = L%16 ; VGPR j holds M = j + 8*(L/16)

__device__ inline v8f wmma_gemm_nn(const float* __restrict__ A, int lda,
                                   const float* __restrict__ B, int ldb,
                                   int K, v8f c) {
  const int lane = (int)(threadIdx.x & 31u);
  const int half = lane >> 4;
  const int l16  = lane & 15;
  const float* arow = A + (size_t)l16 * lda;
  for (int k = 0; k < K; k += 4) {
    v2f a, b;
    a.x = arow[k + 2 * half];
    a.y = arow[k + 2 * half + 1];
    b.x = B[(size_t)(k + half) * ldb + l16];
    b.y = B[(size_t)(k + 2 + half) * ldb + l16];
    c = __builtin_amdgcn_wmma_f32_16x16x4_f32(false, a, false, b,
                                              (short)0, c, false, false);
  }
  return c;
}

__device__ inline v8f wmma_gemm_nt(const float* __restrict__ A, int lda,
                                   const float* __restrict__ Bt, int ldb,
                                   int K, v8f c) {
  const int lane = (int)(threadIdx.x & 31u);
  const int half = lane >> 4;
  const int l16  = lane & 15;
  const float* arow = A + (size_t)l16 * lda;
  const float* brow = Bt + (size_t)l16 * ldb;
  for (int k = 0; k < K; k += 4) {
    v2f a, b;
    a.x = arow[k + 2 * half];
    a.y = arow[k + 2 * half + 1];
    b.x = brow[k + half];
    b.y = brow[k + 2 + half];
    c = __builtin_amdgcn_wmma_f32_16x16x4_f32(false, a, false, b,
                                              (short)0, c, false, false);
  }
  return c;
}

__device__ inline void wmma_store(float* __restrict__ C, int ldc, v8f c,
                                  float scale, float bias) {
  const int lane = (int)(threadIdx.x & 31u);
  const int half = lane >> 4;
  const int l16  = lane & 15;
#pragma unroll
  for (int j = 0; j < 8; ++j)
    C[(size_t)(j + 8 * half) * ldc + l16] = c[j] * scale + bias;
}

__device__ inline int wave_id() {
  return (int)blockIdx.x * (int)(blockDim.x >> 5) + (int)(threadIdx.x >> 5);
}

// ---- Kernel 1: per-head QKV projections ------------------------------------
// qh/kh/vh layout: (h, rows=4096, 64), rows = b*2048 + l
__global__ void k_proj_qkv(const float* __restrict__ q, const float* __restrict__ k,
                           const float* __restrict__ v, const float* __restrict__ wq,
                           const float* __restrict__ wk, const float* __restrict__ wv,
                           float* __restrict__ qh, float* __restrict__ kh,
                           float* __restrict__ vh) {
  int w = wave_id();                    // 3*8*256*4 = 24576 waves
  const int tn = w & 3;  w >>= 2;
  const int tm = w & 255; w >>= 8;
  const int h  = w & 7;  w >>= 3;
  const int t  = w;

  const float* X = (t == 0) ? q : ((t == 1) ? k : v);
  const float* W = (t == 0) ? wq : ((t == 1) ? wk : wv);
  float*       O = (t == 0) ? qh : ((t == 1) ? kh : vh);

  v8f c = {};
  c = wmma_gemm_nn(X + (size_t)tm * 16 * DM_, DM_,
                   W + (size_t)h * DM_ * DK_ + tn * 16, DK_, DM_, c);
  wmma_store(O + ((size_t)h * ROWS_ + (size_t)tm * 16) * DK_ + tn * 16,
             DK_, c, 1.0f, 0.0f);
}

// ---- Kernel 2: fused scores -> softmax -> top-1 mask -> attn.V -------------
// One block (8 waves, 256 threads) per (slice, 16-row tile). Grid = 16*128.
__global__ void k_fused_attn(const float* __restrict__ qh, const float* __restrict__ kh,
                             const float* __restrict__ vh,
                             float* __restrict__ attn, float* __restrict__ cat) {
  extern __shared__ float smem[];
  float* stripe  = smem;                       // 16 x STRIPE_LD
  float* red     = smem + 16 * STRIPE_LD;      // 256
  float* partial = red + 256;                  // 4 tiles x 256

  const int slice = (int)blockIdx.x >> 7;      // h*2 + b
  const int tm    = (int)blockIdx.x & 127;
  const int h = slice >> 1, b = slice & 1;

  const int tid  = (int)threadIdx.x;
  const int w    = tid >> 5;
  const int lane = tid & 31;
  const int half = lane >> 4, l16 = lane & 15;

  const float* Aq = qh + ((size_t)h * ROWS_ + (size_t)b * LQ_ + (size_t)tm * 16) * DK_;
  const float* Kt = kh + ((size_t)h * ROWS_ + (size_t)b * LQ_) * DK_;
  const float* Vv = vh + ((size_t)h * ROWS_ + (size_t)b * LQ_) * DK_;

  // --- phase 1: scores stripe (16 x 2048) into LDS, scaled -------------------
  for (int i = 0; i < 16; ++i) {
    const int tn = w * 16 + i;                 // 8 waves x 16 tiles = 128
    v8f c = {};
    c = wmma_gemm_nt(Aq, DK_, Kt + (size_t)tn * 16 * DK_, DK_, DK_, c);
#pragma unroll
    for (int j = 0; j < 8; ++j)
      stripe[(j + 8 * half) * STRIPE_LD + tn * 16 + l16] = c[j] * INV_TEMPER;
  }
  __syncthreads();

  // --- phase 2: row softmax + hard top-1 mask in LDS -------------------------
  // 16 threads per row; thread covers cols { cbase + 16*i : i < 128 }
  const int r = tid >> 4, cbase = tid & 15;
  float* srow = stripe + r * STRIPE_LD;

  float m = -3.402823466e+38f;
#pragma unroll 8
  for (int i = 0; i < 128; ++i) m = fmaxf(m, srow[cbase + i * 16]);
  red[tid] = m; __syncthreads();
  for (int st = 8; st > 0; st >>= 1) {
    if (cbase < st) red[tid] = fmaxf(red[tid], red[tid + st]);
    __syncthreads();
  }
  m = red[r << 4]; __syncthreads();

  float sum = 0.0f;
#pragma unroll 8
  for (int i = 0; i < 128; ++i) {
    float e = __expf(srow[cbase + i * 16] - m);
    srow[cbase + i * 16] = e;
    sum += e;
  }
  red[tid] = sum; __syncthreads();
  for (int st = 8; st > 0; st >>= 1) {
    if (cbase < st) red[tid] += red[tid + st];
    __syncthreads();
  }
  const float invZ = 1.0f / red[r << 4]; __syncthreads();

  float am = 0.0f;
#pragma unroll 8
  for (int i = 0; i < 128; ++i) {
    float a = srow[cbase + i * 16] * invZ;
    srow[cbase + i * 16] = a;
    am = fmaxf(am, a);
  }
  red[tid] = am; __syncthreads();
  for (int st = 8; st > 0; st >>= 1) {
    if (cbase < st) red[tid] = fmaxf(red[tid], red[tid + st]);
    __syncthreads();
  }
  am = red[r << 4]; __syncthreads();

  // mask in LDS + single coalesced write of attn to global
  float* grow = attn + ((size_t)slice * LQ_ + (size_t)tm * 16 + r) * LQ_;
#pragma unroll 8
  for (int i = 0; i < 128; ++i) {
    const int col = cbase + i * 16;
    float a = srow[col];
    a = (a == am) ? a : 0.0f;
    srow[col] = a;
    grow[col] = a;
  }
  __syncthreads();

  // --- phase 3: attn.V from LDS stripe, split-K across waves -----------------
  // wave w: column tile tn = w&3 (64/16), K half = w>>2 (2 x 1024)
  const int tn = w & 3;
  const int k0 = (w >> 2) * 1024;
  const float* Bv = Vv + tn * 16;
  v8f c = {};
  for (int k = k0; k < k0 + 1024; k += 4) {
    v2f a, bb;
    a.x = stripe[l16 * STRIPE_LD + k + 2 * half];
    a.y = stripe[l16 * STRIPE_LD + k + 2 * half + 1];
    bb.x = Bv[(size_t)(k + half) * DK_ + l16];
    bb.y = Bv[(size_t)(k + 2 + half) * DK_ + l16];
    c = __builtin_amdgcn_wmma_f32_16x16x4_f32(false, a, false, bb,
                                              (short)0, c, false, false);
  }
  if (w >= 4) {
#pragma unroll
    for (int j = 0; j < 8; ++j)
      partial[tn * 256 + (j + 8 * half) * 16 + l16] = c[j];
  }
  __syncthreads();
  if (w < 4) {
    // concat layout: row = b*LQ + q, col = h*64 + n
    float* O = cat + ((size_t)b * LQ_ + (size_t)tm * 16) * DM_ + (size_t)h * DK_ + tn * 16;
#pragma unroll
    for (int j = 0; j < 8; ++j)
      O[(size_t)(j + 8 * half) * DM_ + l16] =
          c[j] + partial[tn * 256 + (j + 8 * half) * 16 + l16];
  }
}

// ---- Kernel 3: output projection: tmp = cat . proj_w^T + proj_b ------------
__global__ void k_proj_out(const float* __restrict__ cat, const float* __restrict__ pw,
                           const float* __restrict__ pb, float* __restrict__ tmp) {
  int w = wave_id();                    // 256 * 32 = 8192 waves
  const int tn = w & 31; w >>= 5;
  const int tm = w;

  v8f c = {};
  c = wmma_gemm_nt(cat + (size_t)tm * 16 * DM_, DM_,
                   pw + (size_t)tn * 16 * DM_, DM_, DM_, c);
  const float bias = pb[tn * 16 + (int)(threadIdx.x & 15u)];
  wmma_store(tmp + (size_t)tm * 16 * DM_ + tn * 16, DM_, c, 1.0f, bias);
}

// ---- Kernel 4: out = [q, tmp] . lin_w^T + lin_b ----------------------------
__global__ void k_final(const float* __restrict__ q, const float* __restrict__ tmp,
                        const float* __restrict__ lw, const float* __restrict__ lb,
                        float* __restrict__ out) {
  int w = wave_id();                    // 256 * 32 = 8192 waves
  const int tn = w & 31; w >>= 5;
  const int tm = w;

  v8f c = {};
  c = wmma_gemm_nt(q   + (size_t)tm * 16 * DM_, DM_,
                   lw  + (size_t)tn * 16 * (2 * DM_),       2 * DM_, DM_, c);
  c = wmma_gemm_nt(tmp + (size_t)tm * 16 * DM_, DM_,
                   lw  + (size_t)tn * 16 * (2 * DM_) + DM_, 2 * DM_, DM_, c);
  const float bias = lb[tn * 16 + (int)(threadIdx.x & 15u)];
  wmma_store(out + (size_t)tm * 16 * DM_ + tn * 16, DM_, c, 1.0f, bias);
}

// ---------------------------------------------------------------------------
extern "C" void kernel_launch(void* const* d_in, const int* in_sizes, int n_in,
                              void* d_out, int out_size, void* d_ws, size_t ws_size,
                              hipStream_t stream) {
  const float* q      = (const float*)d_in[0];
  const float* k      = (const float*)d_in[1];
  const float* v      = (const float*)d_in[2];
  const float* w_qs   = (const float*)d_in[3];
  const float* w_ks   = (const float*)d_in[4];
  const float* w_vs   = (const float*)d_in[5];
  const float* proj_w = (const float*)d_in[6];
  const float* proj_b = (const float*)d_in[7];
  const float* lin_w  = (const float*)d_in[8];
  const float* lin_b  = (const float*)d_in[9];

  float* out  = (float*)d_out;                       // (2, 2048, 512)
  float* attn = out + (size_t)ROWS_ * DM_;           // (16, 2048, 2048)

  float* ws  = (float*)d_ws;
  const size_t HSZ = (size_t)NH_ * ROWS_ * DK_;      // 2,097,152 floats
  float* qh  = ws;
  float* kh  = qh + HSZ;
  float* vh  = kh + HSZ;
  float* cat = vh + HSZ;                             // (4096, 512)
  float* tmp = cat + (size_t)ROWS_ * DM_;            // (4096, 512)

  const int BLOCK = 256;                             // 8 wave32 waves

  // allow >64KB dynamic LDS for the fused kernel (gfx1250: 320KB/WGP)
  (void)hipFuncSetAttribute((const void*)k_fused_attn,
                            hipFuncAttributeMaxDynamicSharedMemorySize,
                            SMEM_BYTES);

  // 1) per-head QKV projections
  k_proj_qkv<<<24576 / 8, BLOCK, 0, stream>>>(q, k, v, w_qs, w_ks, w_vs, qh, kh, vh);

  // 2) fused attention: 16 slices x 128 row tiles
  k_fused_attn<<<16 * 128, BLOCK, SMEM_BYTES, stream>>>(qh, kh, vh, attn, cat);

  // 3) output projection
  k_proj_out<<<8192 / 8, BLOCK, 0, stream>>>(cat, proj_w, proj_b, tmp);

  // 4) final linear on [q, tmp]
  k_final<<<8192 / 8, BLOCK, 0, stream>>>(q, tmp, lin_w, lin_b, out);
}